// LookUpGCN_29437705847415
// MI455X (gfx1250) — compile-verified
//
#include <hip/hip_runtime.h>
#include <hip/hip_bf16.h>

typedef float v2f __attribute__((ext_vector_type(2)));
typedef float v8f __attribute__((ext_vector_type(8)));

#define DIM 128
#define LN_EPS 1e-5f
// K-pair interleaved W staging: 64 K-pairs, each 128 cols x 2 floats, pitch 288
// (288 mod 64 == 32 -> the two lane-halves read disjoint bank sets, b64-conflict-free)
#define KP_PITCH 288

// ---------- degree / normalization ----------
__global__ void k_deg_init(float* __restrict__ deg, int N) {
  int i = blockIdx.x * blockDim.x + threadIdx.x;
  if (i < N) deg[i] = 1.0f;  // self-loop contributes 1
}

__global__ void k_deg_accum(const int* __restrict__ dst, float* __restrict__ deg, int E) {
  int e = blockIdx.x * blockDim.x + threadIdx.x;
  if (e < E) unsafeAtomicAdd(&deg[dst[e]], 1.0f);
}

__global__ void k_deg_rsqrt(float* __restrict__ deg, int N) {
  int i = blockIdx.x * blockDim.x + threadIdx.x;
  if (i < N) deg[i] = rsqrtf(deg[i]);
}

// ---------- embedding gather: x[i] = emb[ids[i]] ----------
__global__ void k_lookup(const int* __restrict__ ids, const float* __restrict__ emb,
                         float* __restrict__ x, int N) {
  int gid = blockIdx.x * blockDim.x + threadIdx.x;
  int i = gid >> 5, j = gid & 31;
  if (i >= N) return;
  ((float4*)x)[(size_t)i * 32 + j] = ((const float4*)emb)[(size_t)ids[i] * 32 + j];
}

// ---------- h = x @ W via v_wmma_f32_16x16x4_f32 ----------
// 8 waves/block; wave computes 16 rows x 128 cols = 8 accumulators of 16x16.
__global__ void __launch_bounds__(256) k_gemm_wmma(const float* __restrict__ x,
                                                   const float* __restrict__ W,
                                                   float* __restrict__ h, int N) {
  __shared__ float lw[64 * KP_PITCH];  // 72 KB: W with K-pairs element-interleaved
  // lw[kp*KP_PITCH + c*2 + p] = W[2*kp + p][c]
  for (int i = threadIdx.x; i < DIM * 32; i += 256) {
    int k = i >> 5, c4 = i & 31;
    float4 wv = *(const float4*)&W[k * DIM + c4 * 4];
    float* p = &lw[(k >> 1) * KP_PITCH + (c4 * 4) * 2 + (k & 1)];
    p[0] = wv.x; p[2] = wv.y; p[4] = wv.z; p[6] = wv.w;
  }
  __syncthreads();

  int wave = threadIdx.x >> 5;
  int lane = threadIdx.x & 31;
  int tile = blockIdx.x * 8 + wave;
  int row0 = tile * 16;
  if (row0 >= N) return;                 // wave-uniform exit

  int m  = lane & 15;                    // A row / B col / D col (ISA 7.12.2)
  int kh = lane >> 4;                    // lane-half selects K pair {0,1} vs {2,3}

  v8f acc[8] = {};
  for (int k0 = 0; k0 < DIM; k0 += 4) {
    int k  = k0 + kh * 2;
    int kp = (k0 >> 1) + kh;                           // which interleaved K-pair
    int r = row0 + m; r = (r < N) ? r : (N - 1);       // branchless clamp: EXEC stays all-ones
    v2f a = *(const v2f*)&x[(size_t)r * DIM + k];      // A[m][k], A[m][k+1]
#pragma unroll
    for (int t = 0; t < 8; ++t) {
      // single aligned b64 LDS load: {W[k][n], W[k+1][n]}
      v2f b = *(const v2f*)&lw[kp * KP_PITCH + (t * 16 + m) * 2];
      acc[t] = __builtin_amdgcn_wmma_f32_16x16x4_f32(
          false, a, false, b, (short)0, acc[t], false, false);
    }
  }
  // D layout: vgpr r, lanes0-15 -> M=r, lanes16-31 -> M=r+8; N = lane&15
#pragma unroll
  for (int t = 0; t < 8; ++t) {
#pragma unroll
    for (int r = 0; r < 8; ++r) {
      int row = row0 + r + kh * 8;
      if (row < N) h[(size_t)row * DIM + t * 16 + m] = acc[t][r];
    }
  }
}

// ---------- agg init with self-loop term: agg[i] = dinv[i]^2 * h[i] ----------
__global__ void k_selfloop(const float* __restrict__ dinv, const float* __restrict__ h,
                           float* __restrict__ agg, int N) {
  int gid = blockIdx.x * blockDim.x + threadIdx.x;
  int i = gid >> 5, j = gid & 31;
  if (i >= N) return;
  float s = dinv[i]; s = s * s;
  float4 v = ((const float4*)h)[(size_t)i * 32 + j];
  v.x *= s; v.y *= s; v.z *= s; v.w *= s;
  ((float4*)agg)[(size_t)i * 32 + j] = v;
}

// ---------- edge scatter: one wave per edge, float4 per lane, fp32 L2 atomics ----------
__global__ void k_edge_scatter(const int* __restrict__ src, const int* __restrict__ dst,
                               const float* __restrict__ dinv, const float* __restrict__ h,
                               float* __restrict__ agg, int E) {
  unsigned gid = blockIdx.x * blockDim.x + threadIdx.x;
  int e = (int)(gid >> 5), lane = (int)(gid & 31);
  if (e >= E) return;
  int s = src[e], d = dst[e];
  float nrm = dinv[s] * dinv[d];
  float4 v = *(const float4*)&h[(size_t)s * DIM + lane * 4];
  float* p = &agg[(size_t)d * DIM + lane * 4];
  unsafeAtomicAdd(p + 0, nrm * v.x);
  unsafeAtomicAdd(p + 1, nrm * v.y);
  unsafeAtomicAdd(p + 2, nrm * v.z);
  unsafeAtomicAdd(p + 3, nrm * v.w);
}

// ---------- xout = LayerNorm(xin + agg + bias) * gamma + beta, one wave/node ----------
__global__ void k_resid_ln(const float* __restrict__ xin, const float* __restrict__ agg,
                           const float* __restrict__ bias, const float* __restrict__ gamma,
                           const float* __restrict__ beta, float* __restrict__ xout, int N) {
  int gid = blockIdx.x * blockDim.x + threadIdx.x;
  int i = gid >> 5, j = gid & 31;
  if (i >= N) return;
  float4 xv = ((const float4*)xin)[(size_t)i * 32 + j];
  float4 av = ((const float4*)agg)[(size_t)i * 32 + j];
  float4 bv = ((const float4*)bias)[j];
  float4 v;
  v.x = xv.x + av.x + bv.x; v.y = xv.y + av.y + bv.y;
  v.z = xv.z + av.z + bv.z; v.w = xv.w + av.w + bv.w;

  float s = v.x + v.y + v.z + v.w;
#pragma unroll
  for (int o = 16; o >= 1; o >>= 1) s += __shfl_xor(s, o, 32);
  float mu = s * (1.0f / 128.0f);

  float4 c;
  c.x = v.x - mu; c.y = v.y - mu; c.z = v.z - mu; c.w = v.w - mu;
  float q = c.x * c.x + c.y * c.y + c.z * c.z + c.w * c.w;
#pragma unroll
  for (int o = 16; o >= 1; o >>= 1) q += __shfl_xor(q, o, 32);
  float is = rsqrtf(q * (1.0f / 128.0f) + LN_EPS);

  float4 gv = ((const float4*)gamma)[j];
  float4 ev = ((const float4*)beta)[j];
  float4 o4;
  o4.x = c.x * is * gv.x + ev.x; o4.y = c.y * is * gv.y + ev.y;
  o4.z = c.z * is * gv.z + ev.z; o4.w = c.w * is * gv.w + ev.w;
  ((float4*)xout)[(size_t)i * 32 + j] = o4;
}

extern "C" void kernel_launch(void* const* d_in, const int* in_sizes, int n_in,
                              void* d_out, int out_size, void* d_ws, size_t ws_size,
                              hipStream_t stream) {
  const int*   node_ids = (const int*)d_in[0];
  const int*   eidx     = (const int*)d_in[1];
  const float* emb      = (const float*)d_in[2];
  const float* W1  = (const float*)d_in[3];
  const float* b1  = (const float*)d_in[4];
  const float* W2  = (const float*)d_in[5];
  const float* b2  = (const float*)d_in[6];
  const float* g1  = (const float*)d_in[7];
  const float* be1 = (const float*)d_in[8];
  const float* g2  = (const float*)d_in[9];
  const float* be2 = (const float*)d_in[10];

  const int N = in_sizes[0];
  const int E = in_sizes[1] / 2;
  const int* src = eidx;
  const int* dst = eidx + E;

  // Workspace layout (floats): dinv[N] | x[N*128] | h[N*128] | agg[N*128]  (~77 MB)
  float* dinv = (float*)d_ws;
  float* x    = dinv + (((size_t)N + 255) & ~(size_t)255);
  float* h    = x + (size_t)N * DIM;
  float* agg  = h + (size_t)N * DIM;
  float* out  = (float*)d_out;

  const int T = 256;
  const int bn   = (N + T - 1) / T;                 // per-node scalar
  const int be_  = (E + T - 1) / T;                 // per-edge scalar
  const int bn32 = (int)(((size_t)N * 32 + T - 1) / T);  // wave-per-node
  const int be32 = (int)(((size_t)E * 32 + T - 1) / T);  // wave-per-edge
  const int bg   = ((N + 15) / 16 + 7) / 8;         // GEMM: 8 row-tiles per block

  // Degree with self-loops -> dinv = deg^{-1/2}
  k_deg_init<<<bn, T, 0, stream>>>(dinv, N);
  k_deg_accum<<<be_, T, 0, stream>>>(dst, dinv, E);
  k_deg_rsqrt<<<bn, T, 0, stream>>>(dinv, N);

  // x = emb[node_ids]
  k_lookup<<<bn32, T, 0, stream>>>(node_ids, emb, x, N);

  // ---- layer 1 ----
  k_gemm_wmma<<<bg, T, 0, stream>>>(x, W1, h, N);
  k_selfloop<<<bn32, T, 0, stream>>>(dinv, h, agg, N);
  k_edge_scatter<<<be32, T, 0, stream>>>(src, dst, dinv, h, agg, E);
  k_resid_ln<<<bn32, T, 0, stream>>>(x, agg, b1, g1, be1, x, N);  // x <- LN(x + agg + b1)

  // ---- layer 2 ----
  k_gemm_wmma<<<bg, T, 0, stream>>>(x, W2, h, N);
  k_selfloop<<<bn32, T, 0, stream>>>(dinv, h, agg, N);
  k_edge_scatter<<<be32, T, 0, stream>>>(src, dst, dinv, h, agg, E);
  k_resid_ln<<<bn32, T, 0, stream>>>(x, agg, b2, g2, be2, out, N);
}